// SpatialGraphConv_16166256902484
// MI455X (gfx1250) — compile-verified
//
#include <hip/hip_runtime.h>

// ---------------------------------------------------------------------------
// SpatialGraphConv for MI455X (gfx1250, wave32).
// - All GEMM stages use V_WMMA_F32_16X16X4_F32 (fp32 in/out): workload is
//   HBM/L2-bandwidth bound (<0.1 flop/byte vs 23.3 TB/s + 192 MB L2), so fp32
//   WMMA saturates the roofline at full reference precision.
// - Propagation stage stages its Y panel into LDS with CDNA5 async copies
//   (GLOBAL_LOAD_ASYNC_TO_LDS_B128 / S_WAIT_ASYNCCNT, ISA ch.10.7/8).
// ---------------------------------------------------------------------------

typedef __attribute__((ext_vector_type(2))) float v2f;
typedef __attribute__((ext_vector_type(8))) float v8f;

__device__ __forceinline__ v8f wmma_f32(v2f a, v2f b, v8f c) {
  // 8 args: (neg_a, A, neg_b, B, c_mod, C, reuse_a, reuse_b)
  return __builtin_amdgcn_wmma_f32_16x16x4_f32(false, a, false, b, (short)0, c,
                                               false, false);
}

constexpr int N_ = 32, C_ = 32, V_ = 400, L_ = 64;
constexpr int VL = V_ * L_;          // 25600
constexpr int NC = N_ * C_;          // 1024
constexpr int ROWS = V_ * V_;        // 160000 MLP rows
constexpr float ALPHA = 0.05f;
constexpr int K_TOP = 20;
constexpr int HSTR = 130;            // padded LDS row stride (bank-conflict free)

// ======================= 1) Edge MLP (64->128->64->2) ======================
// One workgroup = 8 waves, each wave owns 16 rows (128 rows / WG, 1250 WGs).
// Weights live in LDS; inter-layer activations go through a per-wave LDS
// transpose buffer so they can be re-read in WMMA A-fragment layout.
__global__ __launch_bounds__(256) void mlp_kernel(
    const float* __restrict__ adj,   // [ROWS, 64]
    const float* __restrict__ w0, const float* __restrict__ b0,   // 64x128
    const float* __restrict__ w1, const float* __restrict__ b1,   // 128x64
    const float* __restrict__ w2, const float* __restrict__ b2,   // 64x2
    float* __restrict__ glog)        // [2, ROWS]
{
  extern __shared__ float smem[];
  float* sw0 = smem;                 // 64*128
  float* sw1 = sw0 + 64 * 128;       // 128*64
  float* sw2 = sw1 + 128 * 64;       // 128
  float* sb0 = sw2 + 128;            // 128
  float* sb1 = sb0 + 128;            // 64
  float* sb2 = sb1 + 64;             // 2
  float* sh  = sb2 + 2;              // 8 waves * 16 * HSTR

  const int tid = threadIdx.x;
  for (int i = tid; i < 64 * 128; i += 256) sw0[i] = w0[i];
  for (int i = tid; i < 128 * 64; i += 256) sw1[i] = w1[i];
  if (tid < 128) { sw2[tid] = w2[tid]; sb0[tid] = b0[tid]; }
  if (tid < 64) sb1[tid] = b1[tid];
  if (tid < 2) sb2[tid] = b2[tid];
  __syncthreads();

  const int wave = tid >> 5, lane = tid & 31;
  const int m = lane & 15, h = lane >> 4;     // M/N index, K-half
  const int row0 = blockIdx.x * 128 + wave * 16;
  float* hw = sh + wave * 16 * HSTR;          // this wave's [16][HSTR] scratch

  // ---- layer 0: [16,64] @ [64,128], K in 16 steps of 4 ----
  const float* xrow = adj + (size_t)(row0 + m) * 64;
  for (int nt = 0; nt < 8; ++nt) {
    const int n0 = nt * 16;
    v8f acc = {};
#pragma unroll
    for (int k = 0; k < 16; ++k) {
      const int kk = 4 * k + 2 * h;
      v2f a, b;
      a.x = xrow[kk];
      a.y = xrow[kk + 1];
      b.x = sw0[kk * 128 + n0 + m];
      b.y = sw0[(kk + 1) * 128 + n0 + m];
      acc = wmma_f32(a, b, acc);
    }
    const float bias = sb0[n0 + m];
#pragma unroll
    for (int r = 0; r < 8; ++r)                // D: row r+8h, col n0+m
      hw[(r + 8 * h) * HSTR + n0 + m] = fmaxf(acc[r] + bias, 0.f);
  }
  __syncthreads();

  // ---- layer 1: [16,128] @ [128,64], K in 32 steps of 4 ----
  v8f acc1[4];
#pragma unroll
  for (int nt = 0; nt < 4; ++nt) {
    const int n0 = nt * 16;
    v8f acc = {};
#pragma unroll 8
    for (int k = 0; k < 32; ++k) {
      const int kk = 4 * k + 2 * h;
      v2f a, b;
      a.x = hw[m * HSTR + kk];
      a.y = hw[m * HSTR + kk + 1];
      b.x = sw1[kk * 64 + n0 + m];
      b.y = sw1[(kk + 1) * 64 + n0 + m];
      acc = wmma_f32(a, b, acc);
    }
    acc1[nt] = acc;
  }
  __syncthreads();
#pragma unroll
  for (int nt = 0; nt < 4; ++nt) {
    const int n0 = nt * 16;
    const float bias = sb1[n0 + m];
#pragma unroll
    for (int r = 0; r < 8; ++r)
      hw[(r + 8 * h) * HSTR + n0 + m] = fmaxf(acc1[nt][r] + bias, 0.f);
  }
  __syncthreads();

  // ---- layer 2: [16,64] @ [64,2] — N=2, VALU is the right tool.
  // lanes 0..15 -> graph 0, lanes 16..31 -> graph 1; each lane does row m.
  float s = sb2[h];
#pragma unroll 8
  for (int k = 0; k < 64; ++k) s += hw[m * HSTR + k] * sw2[k * 2 + h];
  glog[(size_t)h * ROWS + row0 + m] = s;
}

// ================== 2) softmax + top-20 + renormalize ======================
// One wave32 per (graph,row). Each lane owns <=13 strided columns in regs.
__global__ __launch_bounds__(32) void sparsify_kernel(
    const float* __restrict__ glog, float* __restrict__ Aout)
{
  const int g = blockIdx.x / V_, i = blockIdx.x % V_;
  const float* src = glog + (size_t)g * ROWS + (size_t)i * V_;
  float* dst = Aout + (size_t)g * ROWS + (size_t)i * V_;
  const int lane = threadIdx.x;

  float p[13];
  float mx = -1e30f;
#pragma unroll
  for (int t = 0; t < 13; ++t) {
    const int j = lane + 32 * t;
    p[t] = (j < V_) ? src[j] : -1e30f;
    mx = fmaxf(mx, p[t]);
  }
#pragma unroll
  for (int s = 16; s > 0; s >>= 1) mx = fmaxf(mx, __shfl_xor(mx, s, 32));

  float sum = 0.f;
#pragma unroll
  for (int t = 0; t < 13; ++t) {
    const int j = lane + 32 * t;
    p[t] = (j < V_) ? __expf(p[t] - mx) : 0.f;
    sum += p[t];
  }
#pragma unroll
  for (int s = 16; s > 0; s >>= 1) sum += __shfl_xor(sum, s, 32);
  const float invs = 1.f / sum;
#pragma unroll
  for (int t = 0; t < 13; ++t) p[t] *= invs;

  // 20 rounds of cross-lane argmax
  float work[13];
  unsigned keep = 0u;
#pragma unroll
  for (int t = 0; t < 13; ++t)
    work[t] = (lane + 32 * t < V_) ? p[t] : -1.f;

  float ksum = 0.f;
  for (int it = 0; it < K_TOP; ++it) {
    float bv = -1.f;
    int bi = 0x7fffffff;
#pragma unroll
    for (int t = 0; t < 13; ++t)
      if (work[t] > bv) { bv = work[t]; bi = lane + 32 * t; }
#pragma unroll
    for (int s = 16; s > 0; s >>= 1) {
      const float ov = __shfl_xor(bv, s, 32);
      const int oi = __shfl_xor(bi, s, 32);
      if (ov > bv || (ov == bv && oi < bi)) { bv = ov; bi = oi; }
    }
    ksum += bv;
    if ((bi & 31) == lane) {
      const int t = bi >> 5;
#pragma unroll
      for (int u = 0; u < 13; ++u)
        if (u == t) { work[u] = -1.f; keep |= 1u << u; }
    }
  }
  const float scale = 1.f / (ksum + 1e-8f);
#pragma unroll
  for (int t = 0; t < 13; ++t) {
    const int j = lane + 32 * t;
    if (j < V_) dst[j] = ((keep >> t) & 1u) ? p[t] * scale : 0.f;
  }
}

// ============================ 3) relu(x) ====================================
__global__ __launch_bounds__(256) void relu_kernel(
    const float4* __restrict__ in, float4* __restrict__ out, int n4)
{
  const int i = blockIdx.x * 256 + threadIdx.x;
  if (i < n4) {
    float4 v = in[i];
    v.x = fmaxf(v.x, 0.f); v.y = fmaxf(v.y, 0.f);
    v.z = fmaxf(v.z, 0.f); v.w = fmaxf(v.w, 0.f);
    out[i] = v;
  }
}

// ===================== 4) propagation: a*x + (1-a)*A^T y ====================
// One workgroup per (n,c). The 400x64 fp32 Y panel (100 KB) is staged into
// LDS ONCE with CDNA5 async global->LDS copies (ASYNCcnt-tracked), then the
// block's 8 waves compute all 100 16x16 output tiles (K=400 in 100 WMMA
// steps of K=4) with B fragments fed from LDS. A (2x400x400, 1.3 MB) stays
// L2-resident; global_prefetch_b8 warms each tile's A panel.
__global__ __launch_bounds__(256) void prop_kernel(
    const float* __restrict__ Yin, const float* __restrict__ X0,
    const float* __restrict__ Ag,  // [V][V], A[v][w]
    float* __restrict__ Yout)
{
  extern __shared__ float sy[];               // VL floats = 100 KB
  const int nc = blockIdx.x;
  const float* y = Yin + (size_t)nc * VL;
  const int tid = threadIdx.x;

  // ---- async stage: 256 lanes x b128 x 25 iters = 102400 B ----
  for (int i = tid * 4; i < VL; i += 1024) {
    const unsigned lds = (unsigned)(unsigned long long)(sy + i);   // LDS offset
    const unsigned long long ga = (unsigned long long)(y + i);
    asm volatile("global_load_async_to_lds_b128 %0, %1, off"
                 :: "v"(lds), "v"(ga) : "memory");
  }
  asm volatile("s_wait_asynccnt 0x0" ::: "memory");
  __syncthreads();

  const int wave = tid >> 5, lane = tid & 31;
  const int m = lane & 15, h = lane >> 4;
  const float* x = X0 + (size_t)nc * VL;
  float* o = Yout + (size_t)nc * VL;

  for (int t = wave; t < 100; t += 8) {
    const int w0 = (t >> 2) * 16;             // 25 w-tiles
    const int l0 = (t & 3) * 16;              // 4 l-tiles
    __builtin_prefetch(Ag + (size_t)(2 * h) * V_ + w0 + m, 0, 1);
    v8f acc = {};
#pragma unroll 4
    for (int k = 0; k < 100; ++k) {
      const int v0 = 4 * k + 2 * h;
      v2f a, b;
      a.x = Ag[(size_t)v0 * V_ + w0 + m];     // At[w][v] = A[v][w]
      a.y = Ag[(size_t)(v0 + 1) * V_ + w0 + m];
      b.x = sy[v0 * L_ + l0 + m];
      b.y = sy[(v0 + 1) * L_ + l0 + m];
      acc = wmma_f32(a, b, acc);
    }
#pragma unroll
    for (int r = 0; r < 8; ++r) {
      const int idx = (w0 + r + 8 * h) * L_ + l0 + m;
      o[idx] = ALPHA * x[idx] + (1.f - ALPHA) * acc[r];
    }
  }
}

// ========================= 5) 1x1 conv (channel mix) ========================
// Per n: out[32,25600] = W[32,160] @ cat[160,25600]; W staged in padded LDS.
__global__ __launch_bounds__(256) void conv_kernel(
    const float* __restrict__ X0, const float* __restrict__ C1,
    const float* __restrict__ C2, const float* __restrict__ C3,
    const float* __restrict__ C4,
    const float* __restrict__ convw,  // [32][160]
    const float* __restrict__ convb,  // [32]
    float* __restrict__ out)          // [N,32,VL] pre-BN
{
  __shared__ float sw[32 * 162];      // padded stride 162
  __shared__ float sb[32];
  const int tid = threadIdx.x;
  for (int i = tid; i < 32 * 160; i += 256)
    sw[(i / 160) * 162 + (i % 160)] = convw[i];
  if (tid < 32) sb[tid] = convb[tid];
  __syncthreads();

  const float* chunks[5] = {X0, C1, C2, C3, C4};
  const int wave = tid >> 5, lane = tid & 31;
  const int m = lane & 15, h = lane >> 4;
  const int t = blockIdx.x * 8 + wave;
  const int n = t / 3200;
  const int rem = t % 3200;
  const int o0 = (rem / 1600) * 16;        // 2 o-tiles
  const int vl0 = (rem % 1600) * 16;       // 1600 vl-tiles

  v8f acc = {};
#pragma unroll 4
  for (int k = 0; k < 40; ++k) {
    const int c0 = 4 * k + 2 * h;
    v2f a, b;
    a.x = sw[(o0 + m) * 162 + c0];
    a.y = sw[(o0 + m) * 162 + c0 + 1];
    const float* p0 = chunks[c0 >> 5] + (size_t)(n * 32 + (c0 & 31)) * VL;
    const float* p1 = chunks[(c0 + 1) >> 5] + (size_t)(n * 32 + ((c0 + 1) & 31)) * VL;
    b.x = p0[vl0 + m];
    b.y = p1[vl0 + m];
    acc = wmma_f32(a, b, acc);
  }
#pragma unroll
  for (int r = 0; r < 8; ++r) {
    const int o = o0 + r + 8 * h;
    out[((size_t)n * 32 + o) * VL + vl0 + m] = acc[r] + sb[o];
  }
}

// ===================== 6) BatchNorm (deterministic 2-stage) =================
__global__ __launch_bounds__(256) void bn_stats_kernel(
    const float* __restrict__ out, float* __restrict__ ps, float* __restrict__ pq)
{
  const int n = blockIdx.x >> 5, o = blockIdx.x & 31;
  const float* p = out + ((size_t)n * 32 + o) * VL;
  float s = 0.f, q = 0.f;
  for (int i = threadIdx.x; i < VL; i += 256) { const float v = p[i]; s += v; q += v * v; }
#pragma unroll
  for (int sh = 16; sh > 0; sh >>= 1) { s += __shfl_xor(s, sh, 32); q += __shfl_xor(q, sh, 32); }
  __shared__ float rs[8], rq[8];
  const int wave = threadIdx.x >> 5, lane = threadIdx.x & 31;
  if (lane == 0) { rs[wave] = s; rq[wave] = q; }
  __syncthreads();
  if (threadIdx.x == 0) {
    float a = 0.f, b = 0.f;
#pragma unroll
    for (int i = 0; i < 8; ++i) { a += rs[i]; b += rq[i]; }
    ps[blockIdx.x] = a; pq[blockIdx.x] = b;
  }
}

__global__ void bn_finalize_kernel(const float* __restrict__ ps,
                                   const float* __restrict__ pq,
                                   float* __restrict__ mv)
{
  const int o = threadIdx.x;  // 32 threads
  float s = 0.f, q = 0.f;
  for (int n = 0; n < 32; ++n) { s += ps[n * 32 + o]; q += pq[n * 32 + o]; }
  const float inv_cnt = 1.f / (float)(N_ * VL);
  const float mean = s * inv_cnt;
  const float var = q * inv_cnt - mean * mean;   // biased, as jnp.var
  mv[o] = mean;
  mv[32 + o] = rsqrtf(var + 1e-5f);
}

__global__ __launch_bounds__(256) void bn_apply_kernel(
    float* __restrict__ out, const float* __restrict__ mv,
    const float* __restrict__ gamma, const float* __restrict__ beta)
{
  const size_t i = (size_t)blockIdx.x * 256 + threadIdx.x;
  const int o = (int)((i / VL) & 31);
  const float v = out[i];
  out[i] = (v - mv[o]) * mv[32 + o] * gamma[o] + beta[o];
}

// =============================== launcher ===================================
extern "C" void kernel_launch(void* const* d_in, const int* in_sizes, int n_in,
                              void* d_out, int out_size, void* d_ws, size_t ws_size,
                              hipStream_t stream) {
  const float* inputs = (const float*)d_in[0];
  const float* adj    = (const float*)d_in[1];
  const float* w0 = (const float*)d_in[2];  const float* b0 = (const float*)d_in[3];
  const float* w1 = (const float*)d_in[4];  const float* b1 = (const float*)d_in[5];
  const float* w2 = (const float*)d_in[6];  const float* b2 = (const float*)d_in[7];
  const float* cw = (const float*)d_in[8];  const float* cb = (const float*)d_in[9];
  const float* gm = (const float*)d_in[10]; const float* bt = (const float*)d_in[11];
  float* out = (float*)d_out;

  // workspace layout (floats); ~527 MB total
  float* ws   = (float*)d_ws;
  float* ps   = ws;              // 1024
  float* pq   = ws + 1024;       // 1024
  float* mv   = ws + 2048;       // 64
  float* glog = ws + 2112;       // 2*ROWS
  float* Aadj = glog + 2 * ROWS; // 2*ROWS
  float* X0   = Aadj + 2 * ROWS; // NC*VL
  float* Cb0  = X0  + (size_t)NC * VL;
  float* Cb1  = Cb0 + (size_t)NC * VL;
  float* Cb2  = Cb1 + (size_t)NC * VL;
  float* Cb3  = Cb2 + (size_t)NC * VL;

  constexpr int MLP_SMEM =
      (64 * 128 + 128 * 64 + 128 + 128 + 64 + 2 + 8 * 16 * HSTR) * 4; // ~130 KB
  constexpr int PROP_SMEM = VL * 4;                                   // 100 KB

  mlp_kernel<<<ROWS / 128, 256, MLP_SMEM, stream>>>(adj, w0, b0, w1, b1, w2, b2, glog);
  sparsify_kernel<<<2 * V_, 32, 0, stream>>>(glog, Aadj);
  relu_kernel<<<(NC * VL / 4) / 256, 256, 0, stream>>>(
      (const float4*)inputs, (float4*)X0, NC * VL / 4);

  // y11 = P(A0,x); y12 = P(A0,y11); y21 = P(A1,x); y22 = P(A1,y21)
  prop_kernel<<<NC, 256, PROP_SMEM, stream>>>(X0,  X0, Aadj,        Cb0);
  prop_kernel<<<NC, 256, PROP_SMEM, stream>>>(Cb0, X0, Aadj,        Cb1);
  prop_kernel<<<NC, 256, PROP_SMEM, stream>>>(X0,  X0, Aadj + ROWS, Cb2);
  prop_kernel<<<NC, 256, PROP_SMEM, stream>>>(Cb2, X0, Aadj + ROWS, Cb3);

  conv_kernel<<<N_ * 3200 / 8, 256, 0, stream>>>(X0, Cb0, Cb1, Cb2, Cb3, cw, cb, out);

  bn_stats_kernel<<<N_ * 32, 256, 0, stream>>>(out, ps, pq);
  bn_finalize_kernel<<<1, 32, 0, stream>>>(ps, pq, mv);
  bn_apply_kernel<<<(NC * VL) / 256, 256, 0, stream>>>(out, mv, gm, bt);
}